// Net_35880156790911
// MI455X (gfx1250) — compile-verified
//
#include <hip/hip_runtime.h>
#include <stdint.h>

#define N_NODES 100000
#define N_EDGES 3200000
#define D_FEAT  512
#define H1      16
#define H2      2
#define TILES   (N_NODES / 16)   // 6250, exact

typedef __attribute__((ext_vector_type(2))) float v2f;
typedef __attribute__((ext_vector_type(8))) float v8f;

// ---------------------------------------------------------------- degrees ---
__global__ void __launch_bounds__(256) k_init_deg(uint32_t* deg) {
    int i = blockIdx.x * 256 + threadIdx.x;
    if (i < N_NODES) deg[i] = 1u;            // self-loop contributes 1
}

__global__ void __launch_bounds__(256) k_count_deg(const long long* __restrict__ ei,
                                                   uint32_t* deg) {
    int e = blockIdx.x * 256 + threadIdx.x;
    if (e < N_EDGES) atomicAdd(&deg[(int)ei[e]], 1u);   // row = edge_index[0]
}

__global__ void __launch_bounds__(256) k_isq(const uint32_t* __restrict__ deg,
                                             float* __restrict__ isq) {
    int i = blockIdx.x * 256 + threadIdx.x;
    if (i < N_NODES) isq[i] = rsqrtf((float)deg[i]);    // deg >= 1 always
}

// ------------------------------------------------- GEMM1: h1 = x @ W1 -------
// One wave per 16x16 output tile. K accumulated 4 at a time with
// V_WMMA_F32_16X16X4_F32 (exact fp32). W1 staged transposed in LDS.
#define LDS_STRIDE 516   // 516 % 64 == 4 -> 16 lanes hit 16 distinct banks

__global__ void __launch_bounds__(256) k_gemm1(const float* __restrict__ x,
                                               const float* __restrict__ W1,
                                               float* __restrict__ h1) {
    __shared__ float sWt[H1 * LDS_STRIDE];
    for (int i = threadIdx.x; i < D_FEAT * H1; i += 256) {
        int k = i >> 4;          // 0..511
        int n = i & 15;          // 0..15
        sWt[n * LDS_STRIDE + k] = W1[i];   // W1 is [k][n] row-major
    }
    __syncthreads();

    int wave = threadIdx.x >> 5;
    int lane = threadIdx.x & 31;
    int tile = blockIdx.x * 8 + wave;
    if (tile >= TILES) return;               // wave-uniform; EXEC stays all-1s

    int m   = tile * 16 + (lane & 15);       // node row for A
    int kh  = (lane >> 4) << 1;              // lanes 0-15: K+{0,1}; 16-31: K+{2,3}
    int n   = lane & 15;                     // output column for B

    const float* xrow = x   + (size_t)m * D_FEAT;
    const float* wrow = sWt + n * LDS_STRIDE;

    v8f c = {0.f, 0.f, 0.f, 0.f, 0.f, 0.f, 0.f, 0.f};
#pragma unroll 8
    for (int k0 = 0; k0 < D_FEAT; k0 += 4) {
        v2f a = *(const v2f*)(xrow + k0 + kh);   // global_load_b64
        v2f b = *(const v2f*)(wrow + k0 + kh);   // ds_load_b64
        c = __builtin_amdgcn_wmma_f32_16x16x4_f32(
                /*neg_a=*/false, a, /*neg_b=*/false, b,
                /*c_mod=*/(short)0, c, /*reuse_a=*/false, /*reuse_b=*/false);
    }

    // C/D layout: VGPR v, lanes 0-15 -> M=v, lanes 16-31 -> M=v+8; N=lane&15
    int rbase = tile * 16 + (lane >> 4) * 8;
    int col   = lane & 15;
#pragma unroll
    for (int v = 0; v < 8; ++v)
        h1[(size_t)(rbase + v) * H1 + col] = c[v];
}

// ---------------------------------------------- layer-1 scatter-aggregate ---
__global__ void __launch_bounds__(256) k_selfloop1(const float* __restrict__ h1,
                                                   const float* __restrict__ isq,
                                                   float* __restrict__ agg) {
    int t = blockIdx.x * 256 + threadIdx.x;          // over N*16
    if (t < N_NODES * H1) {
        float s = isq[t >> 4];
        agg[t] = h1[t] * s * s;                      // init + self-loop term
    }
}

__global__ void __launch_bounds__(256) k_scatter1(const long long* __restrict__ ei,
                                                  const float* __restrict__ h1,
                                                  const float* __restrict__ isq,
                                                  float* __restrict__ agg) {
    int t = blockIdx.x * 256 + threadIdx.x;          // 4 threads per edge
    if (t >= N_EDGES * 4) return;
    int e  = t >> 2;
    int c4 = (t & 3) << 2;
    int r  = (int)ei[e];                             // scatter target (row)
    int cc = (int)ei[N_EDGES + e];                   // gather source  (col)
    float nrm = isq[r] * isq[cc];
    const float* src = h1  + (size_t)cc * H1 + c4;
    float*       dst = agg + (size_t)r  * H1 + c4;
#pragma unroll
    for (int j = 0; j < 4; ++j)
        atomicAdd(dst + j, src[j] * nrm);
}

// ------------------------------- bias+ReLU, tiny GEMM2, self-loop of out ----
__global__ void __launch_bounds__(256) k_layer2(const float* __restrict__ agg,
                                                const float* __restrict__ b1,
                                                const float* __restrict__ W2,
                                                const float* __restrict__ isq,
                                                float* __restrict__ h2,
                                                float* __restrict__ oacc) {
    int nid = blockIdx.x * 256 + threadIdx.x;
    if (nid >= N_NODES) return;
    const float* hrow = agg + (size_t)nid * H1;
    float a0 = 0.f, a1 = 0.f;
#pragma unroll
    for (int j = 0; j < H1; ++j) {
        float v = fmaxf(hrow[j] + b1[j], 0.f);       // bias + ReLU
        a0 = fmaf(v, W2[j * 2 + 0], a0);
        a1 = fmaf(v, W2[j * 2 + 1], a1);
    }
    float s  = isq[nid];
    float s2 = s * s;
    h2[nid * 2 + 0]   = a0;
    h2[nid * 2 + 1]   = a1;
    oacc[nid * 2 + 0] = a0 * s2;                     // init + self-loop term
    oacc[nid * 2 + 1] = a1 * s2;
}

__global__ void __launch_bounds__(256) k_scatter2(const long long* __restrict__ ei,
                                                  const float* __restrict__ h2,
                                                  const float* __restrict__ isq,
                                                  float* __restrict__ oacc) {
    int e = blockIdx.x * 256 + threadIdx.x;
    if (e >= N_EDGES) return;
    int r  = (int)ei[e];
    int cc = (int)ei[N_EDGES + e];
    float nrm = isq[r] * isq[cc];
    atomicAdd(&oacc[r * 2 + 0], h2[cc * 2 + 0] * nrm);
    atomicAdd(&oacc[r * 2 + 1], h2[cc * 2 + 1] * nrm);
}

// ------------------------------------------ bias + 2-class log_softmax -----
__global__ void __launch_bounds__(256) k_final(const float* __restrict__ oacc,
                                               const float* __restrict__ b2,
                                               float* __restrict__ out) {
    int nid = blockIdx.x * 256 + threadIdx.x;
    if (nid >= N_NODES) return;
    float v0 = oacc[nid * 2 + 0] + b2[0];
    float v1 = oacc[nid * 2 + 1] + b2[1];
    float m  = fmaxf(v0, v1);
    float lse = m + logf(expf(v0 - m) + expf(v1 - m));
    out[nid * 2 + 0] = v0 - lse;
    out[nid * 2 + 1] = v1 - lse;
}

// ---------------------------------------------------------------------------
extern "C" void kernel_launch(void* const* d_in, const int* in_sizes, int n_in,
                              void* d_out, int out_size, void* d_ws, size_t ws_size,
                              hipStream_t stream) {
    const float*     x  = (const float*)d_in[0];
    const long long* ei = (const long long*)d_in[1];   // int64 [2, E]
    const float*     W1 = (const float*)d_in[2];
    const float*     b1 = (const float*)d_in[3];
    const float*     W2 = (const float*)d_in[4];
    const float*     b2 = (const float*)d_in[5];
    float* out = (float*)d_out;

    // workspace layout (float offsets, padded/aligned)
    float*    ws   = (float*)d_ws;
    uint32_t* deg  = (uint32_t*)ws;        //  N u32
    float*    isq  = ws + 102400;          //  N f32
    float*    h1   = ws + 204800;          //  N*16
    float*    agg  = ws + 1843200;         //  N*16
    float*    h2   = ws + 3481600;         //  N*2
    float*    oacc = ws + 3686400;         //  N*2   (total ~15.6 MB)

    const int B = 256;
    k_init_deg <<<(N_NODES + B - 1) / B, B, 0, stream>>>(deg);
    k_count_deg<<<(N_EDGES + B - 1) / B, B, 0, stream>>>(ei, deg);
    k_isq      <<<(N_NODES + B - 1) / B, B, 0, stream>>>(deg, isq);

    k_gemm1    <<<(TILES + 7) / 8, B, 0, stream>>>(x, W1, h1);

    k_selfloop1<<<(N_NODES * H1 + B - 1) / B, B, 0, stream>>>(h1, isq, agg);
    k_scatter1 <<<(N_EDGES * 4 + B - 1) / B, B, 0, stream>>>(ei, h1, isq, agg);

    k_layer2   <<<(N_NODES + B - 1) / B, B, 0, stream>>>(agg, b1, W2, isq, h2, oacc);
    k_scatter2 <<<(N_EDGES + B - 1) / B, B, 0, stream>>>(ei, h2, isq, oacc);

    k_final    <<<(N_NODES + B - 1) / B, B, 0, stream>>>(oacc, b2, out);
}